// PartialEncoderEDDI_57767310131605
// MI455X (gfx1250) — compile-verified
//
#include <hip/hip_runtime.h>

typedef float v2f __attribute__((ext_vector_type(2)));
typedef float v8f __attribute__((ext_vector_type(8)));

#define BB 128
#define JJ 4096
#define DD 64
#define HH 128
#define EE 256
#define Z2 64
#define LN_EPS 1e-5f

#define SPLIT 16
#define TILES_PER_BLK ((JJ / 16) / SPLIT) /* 16 tiles of 16 positions = 256 j per block */
#define H1_STRIDE 132                     /* 128 + 4 pad: conflict-free A-frag reads   */
#define P2_STRIDE 68                      /* 64 + 4 pad                                 */

static __device__ __forceinline__ float red8(float v) {
  // reduce across aligned groups of 8 lanes (wave32-safe: masks 1/2/4 stay in-group)
  v += __shfl_xor(v, 1, 32);
  v += __shfl_xor(v, 2, 32);
  v += __shfl_xor(v, 4, 32);
  return v;
}

// ---------------------------------------------------------------------------
// Kernel 0: G[j][h] = hb1[h] + sum_d F[j][d] * hW1[(1+d)*H + h]   (rank-1 prep)
// ---------------------------------------------------------------------------
__global__ __launch_bounds__(128) void k_precompute_G(
    const float* __restrict__ F, const float* __restrict__ hW1,
    const float* __restrict__ hb1, float* __restrict__ G) {
  __shared__ float Fs[DD];
  const int j = blockIdx.x;
  const int tid = threadIdx.x;
  if (tid < DD) Fs[tid] = F[j * DD + tid];
  __syncthreads();
  float acc = hb1[tid];
#pragma unroll 8
  for (int d = 0; d < DD; ++d) acc += Fs[d] * hW1[(1 + d) * HH + tid];
  G[j * HH + tid] = acc;
}

// ---------------------------------------------------------------------------
// Kernel 1: per-(b,j) MLP with fp32 WMMA for layer 2, partial masked sums out.
// Grid: BB*SPLIT blocks of 128 threads (4 waves). Wave w owns N-tile n0=16w.
// ---------------------------------------------------------------------------
__global__ __launch_bounds__(128) void k_main(
    const float* __restrict__ x, const int* __restrict__ mask,
    const float* __restrict__ G, const float* __restrict__ hW1,
    const float* __restrict__ hg1, const float* __restrict__ hbt1,
    const float* __restrict__ hW2, const float* __restrict__ hb2,
    const float* __restrict__ hg2, const float* __restrict__ hbt2,
    float* __restrict__ partial) {
  __shared__ float h1s[16 * H1_STRIDE];
  __shared__ float p2s[16 * P2_STRIDE];
  __shared__ float xs[16];
  __shared__ float mv[16];
  __shared__ float g1s[HH], b1s[HH], g2s[DD], b2s[DD];

  const int tid = threadIdx.x;
  const int b = blockIdx.x >> 4;
  const int split = blockIdx.x & (SPLIT - 1);
  const int jbase = split * (16 * TILES_PER_BLK);

  g1s[tid] = hg1[tid];
  b1s[tid] = hbt1[tid];
  if (tid < DD) { g2s[tid] = hg2[tid]; b2s[tid] = hbt2[tid]; }
  const float w0 = hW1[tid]; // row 0 of hW1 (the x coefficient), col = tid

  const int lane = tid & 31;
  const int n0 = (tid >> 5) * 16;       // wave's N-tile base in [0,64)
  const int nB = n0 + (lane & 15);      // this lane's output column
  const int kb = (lane >> 4) * 2;       // K sub-offset per A/B fragment layout
  const int hi = lane >> 4;

  // Preload B fragments of hW2 (128x64, row-major) into registers: 32 k-steps.
  v2f bf[32];
#pragma unroll
  for (int k = 0; k < 32; ++k) {
    const int row = 4 * k + kb;
    bf[k].x = hW2[row * DD + nB];
    bf[k].y = hW2[(row + 1) * DD + nB];
  }
  const float hb2v = hb2[nB];

  float accd = 0.0f; // threads 0..63: running masked sum for dim d = tid

  for (int t = 0; t < TILES_PER_BLK; ++t) {
    const int j0 = jbase + t * 16;
    if (tid < 16) {
      xs[tid] = x[(size_t)b * JJ + j0 + tid];
      mv[tid] = (mask[(size_t)b * JJ + j0 + tid] > 0) ? 1.0f : 0.0f;
    }
    __syncthreads();

    // pre-act layer 1: rank-1 update, thread owns column h = tid
#pragma unroll
    for (int p = 0; p < 16; ++p)
      h1s[p * H1_STRIDE + tid] = xs[p] * w0 + G[(size_t)(j0 + p) * HH + tid];
    __syncthreads();

    // LN1 + affine + ReLU in place (8 threads per position, 16 cols each)
    {
      const int p = tid >> 3, i = tid & 7;
      float s = 0.f, ss = 0.f, v[16];
#pragma unroll
      for (int q = 0; q < 16; ++q) {
        const float val = h1s[p * H1_STRIDE + i * 16 + q];
        v[q] = val; s += val; ss += val * val;
      }
      s = red8(s); ss = red8(ss);
      const float mean = s * (1.0f / HH);
      const float rstd = rsqrtf(ss * (1.0f / HH) - mean * mean + LN_EPS);
#pragma unroll
      for (int q = 0; q < 16; ++q) {
        const int c = i * 16 + q;
        const float hv = (v[q] - mean) * rstd * g1s[c] + b1s[c];
        h1s[p * H1_STRIDE + c] = fmaxf(hv, 0.0f);
      }
    }
    __syncthreads();

    // Layer 2: (16 x 128) @ (128 x 64) via fp32 WMMA, wave-per-N-tile
    {
      const int m = lane & 15;
      v8f acc = {};
#pragma unroll
      for (int k = 0; k < 32; ++k) {
        v2f a;
        a.x = h1s[m * H1_STRIDE + 4 * k + kb];
        a.y = h1s[m * H1_STRIDE + 4 * k + kb + 1];
        acc = __builtin_amdgcn_wmma_f32_16x16x4_f32(
            false, a, false, bf[k], (short)0, acc, false, false);
      }
#pragma unroll
      for (int vv = 0; vv < 8; ++vv) {
        const int p = vv + 8 * hi; // C/D layout: M = vgpr + 8*(lane>=16)
        p2s[p * P2_STRIDE + nB] = acc[vv] + hb2v;
      }
    }
    __syncthreads();

    // LN2 + affine + ReLU + mask, in place (8 threads per position, 8 cols)
    {
      const int p = tid >> 3, i = tid & 7;
      float s = 0.f, ss = 0.f, v[8];
#pragma unroll
      for (int q = 0; q < 8; ++q) {
        const float val = p2s[p * P2_STRIDE + i * 8 + q];
        v[q] = val; s += val; ss += val * val;
      }
      s = red8(s); ss = red8(ss);
      const float mean = s * (1.0f / DD);
      const float rstd = rsqrtf(ss * (1.0f / DD) - mean * mean + LN_EPS);
      const float mk = mv[p];
#pragma unroll
      for (int q = 0; q < 8; ++q) {
        const int c = i * 8 + q;
        const float hv = (v[q] - mean) * rstd * g2s[c] + b2s[c];
        p2s[p * P2_STRIDE + c] = fmaxf(hv, 0.0f) * mk;
      }
    }
    __syncthreads();

    // deterministic fixed-order accumulation over the 16 positions
    if (tid < DD) {
      float s = 0.f;
#pragma unroll
      for (int p = 0; p < 16; ++p) s += p2s[p * P2_STRIDE + tid];
      accd += s;
    }
    // no barrier needed: p2s is next written only after 3 barriers above
  }
  if (tid < DD) partial[((size_t)b * SPLIT + split) * DD + tid] = accd;
}

// ---------------------------------------------------------------------------
// Kernel 2: finalize pool + 2-layer encoder (tiny). One block per batch row.
// ---------------------------------------------------------------------------
__global__ __launch_bounds__(256) void k_encoder(
    const int* __restrict__ mask, const float* __restrict__ partial,
    const float* __restrict__ eW1, const float* __restrict__ eb1,
    const float* __restrict__ eW2, const float* __restrict__ eb2,
    float* __restrict__ out) {
  __shared__ float red[256];
  __shared__ float cs[DD];
  __shared__ float es[EE];
  const int b = blockIdx.x;
  const int tid = threadIdx.x;

  // observed count
  float cnt = 0.f;
  for (int j = tid; j < JJ; j += 256)
    cnt += (mask[(size_t)b * JJ + j] > 0) ? 1.0f : 0.0f;
  red[tid] = cnt; __syncthreads();
  for (int s = 128; s > 0; s >>= 1) {
    if (tid < s) red[tid] += red[tid + s];
    __syncthreads();
  }
  const float cntf = fmaxf(red[0], 1.0f);
  __syncthreads();

  // mean pool from deterministic partials
  if (tid < DD) {
    float p = 0.f;
    for (int s2 = 0; s2 < SPLIT; ++s2) p += partial[((size_t)b * SPLIT + s2) * DD + tid];
    cs[tid] = p / cntf;
  }
  __syncthreads();

  // FC1: c @ eW1 + eb1, LN (no affine), ReLU
  float a = eb1[tid];
  for (int d = 0; d < DD; ++d) a += cs[d] * eW1[d * EE + tid];
  red[tid] = a; __syncthreads();
  for (int s = 128; s > 0; s >>= 1) { if (tid < s) red[tid] += red[tid + s]; __syncthreads(); }
  const float m1 = red[0] * (1.0f / EE);
  __syncthreads();
  red[tid] = a * a; __syncthreads();
  for (int s = 128; s > 0; s >>= 1) { if (tid < s) red[tid] += red[tid + s]; __syncthreads(); }
  const float v1 = red[0] * (1.0f / EE) - m1 * m1;
  __syncthreads();
  es[tid] = fmaxf((a - m1) * rsqrtf(v1 + LN_EPS), 0.0f);
  __syncthreads();

  // FC2: e @ eW2 + eb2, LN (no affine), ReLU; split -> (mu, logvar)
  float a2 = 0.f;
  if (tid < Z2) {
    a2 = eb2[tid];
    for (int t2 = 0; t2 < EE; ++t2) a2 += es[t2] * eW2[t2 * Z2 + tid];
  }
  red[tid] = (tid < Z2) ? a2 : 0.f; __syncthreads();
  for (int s = 128; s > 0; s >>= 1) { if (tid < s) red[tid] += red[tid + s]; __syncthreads(); }
  const float m2 = red[0] * (1.0f / Z2);
  __syncthreads();
  red[tid] = (tid < Z2) ? a2 * a2 : 0.f; __syncthreads();
  for (int s = 128; s > 0; s >>= 1) { if (tid < s) red[tid] += red[tid + s]; __syncthreads(); }
  const float v2 = red[0] * (1.0f / Z2) - m2 * m2;
  if (tid < Z2) {
    const float o = fmaxf((a2 - m2) * rsqrtf(v2 + LN_EPS), 0.0f);
    if (tid < 32) out[(size_t)b * 32 + tid] = o;                       // mu
    else          out[(size_t)BB * 32 + (size_t)b * 32 + (tid - 32)] = o; // logvar
  }
}

// ---------------------------------------------------------------------------
extern "C" void kernel_launch(void* const* d_in, const int* in_sizes, int n_in,
                              void* d_out, int out_size, void* d_ws, size_t ws_size,
                              hipStream_t stream) {
  const float* x    = (const float*)d_in[0];
  const int*   mask = (const int*)d_in[1];
  const float* F    = (const float*)d_in[2];
  const float* hW1  = (const float*)d_in[3];
  const float* hb1  = (const float*)d_in[4];
  const float* hg1  = (const float*)d_in[5];
  const float* hbt1 = (const float*)d_in[6];
  const float* hW2  = (const float*)d_in[7];
  const float* hb2  = (const float*)d_in[8];
  const float* hg2  = (const float*)d_in[9];
  const float* hbt2 = (const float*)d_in[10];
  const float* eW1  = (const float*)d_in[11];
  const float* eb1  = (const float*)d_in[12];
  const float* eW2  = (const float*)d_in[13];
  const float* eb2  = (const float*)d_in[14];
  float* out = (float*)d_out;

  float* G       = (float*)d_ws;                    // JJ*HH floats (2 MB)
  float* partial = G + (size_t)JJ * HH;             // BB*SPLIT*DD floats (512 KB)

  k_precompute_G<<<JJ, 128, 0, stream>>>(F, hW1, hb1, G);
  k_main<<<BB * SPLIT, 128, 0, stream>>>(x, mask, G, hW1, hg1, hbt1,
                                         hW2, hb2, hg2, hbt2, partial);
  k_encoder<<<BB, 256, 0, stream>>>(mask, partial, eW1, eb1, eW2, eb2, out);
}